// Model_63385127354480
// MI455X (gfx1250) — compile-verified
//
#include <hip/hip_runtime.h>

// ---------------------------------------------------------------------------
// Companion-matrix SSM on MI455X (gfx1250), fp32 end-to-end.
//   Phase 1: Bu = U @ B^T            (WMMA f32 16x16x4)
//   Phase 2: companion scan          (shift + rank-1, LDS double buffer)
//   Phase 3: Y  = X @ C^T + U @ D^T  (WMMA f32 16x16x4)
// Bandwidth-bound (~590 MB traffic, ~9.2 GFLOP) -> fp32 WMMA is the right
// precision: no speedup available from fp16/fp8 and the 2048-step recurrence
// needs fp32 accumulation.
// ---------------------------------------------------------------------------

typedef __attribute__((ext_vector_type(2))) float v2f;
typedef __attribute__((ext_vector_type(8))) float v8f;

#define N_STATE 512
#define M_IN    64
#define P_OUT   64
#define BATCH   32
#define SEQ     2048
#define ROWS    (BATCH * SEQ)   // 65536 rows, r = b*SEQ + t (matches u layout)

// ---------------------------------------------------------------------------
// Phase 1: Bu[r][n] = sum_m u[r][m] * B[n][m]
// One wave computes a 16(M) x 16(N) fp32 tile, K=64 in 16 WMMA steps.
// ---------------------------------------------------------------------------
__global__ __launch_bounds__(256) void ssm_bu_gemm(const float* __restrict__ u,
                                                   const float* __restrict__ Bmat,
                                                   float* __restrict__ bu) {
  const int wave = (blockIdx.x * 256 + threadIdx.x) >> 5;
  const int lane = threadIdx.x & 31;
  const int half = lane >> 4;     // K-pair select (0: K0/K1, 1: K2/K3)
  const int lm   = lane & 15;     // M (for A) / N (for B) within tile

  const int colTiles = N_STATE / 16;          // 32
  const int rowTile  = wave / colTiles;       // 0..4095
  const int colTile  = wave % colTiles;       // 0..31

  const int row = rowTile * 16 + lm;          // output row (A row)
  const int col = colTile * 16 + lm;          // output col (state index n)

  const float* __restrict__ urow = u + (size_t)row * M_IN;
  const float* __restrict__ brow = Bmat + (size_t)col * M_IN;  // B^T[k][n] = B[n][k]

  v8f acc = {};
#pragma unroll
  for (int k0 = 0; k0 < M_IN; k0 += 4) {
    const int ka = k0 + 2 * half;
    v2f av, bv;
    av.x = urow[ka];
    av.y = urow[ka + 1];
    bv.x = brow[ka];
    bv.y = brow[ka + 1];
    acc = __builtin_amdgcn_wmma_f32_16x16x4_f32(false, av, false, bv,
                                                (short)0, acc, false, false);
  }

  // C/D layout: VGPR v -> row (v + 8*half), col lm
  float* __restrict__ out = bu + (size_t)(rowTile * 16) * N_STATE + colTile * 16 + lm;
#pragma unroll
  for (int v = 0; v < 8; ++v)
    out[(size_t)(v + 8 * half) * N_STATE] = acc[v];
}

// ---------------------------------------------------------------------------
// Phase 2: companion scan, one block per batch element, 512 threads.
//   x_t[i] = x_{t-1}[i-1] - a[511-i]*x_{t-1}[511] + bu_t[i]
// Double-buffered LDS -> one barrier per step. Writes x_t in place over Bu_t.
// ---------------------------------------------------------------------------
__global__ __launch_bounds__(512) void ssm_scan(const float* __restrict__ avec,
                                                float* __restrict__ bu) {
  __shared__ float xbuf[2][N_STATE];
  const int i = threadIdx.x;        // state index
  const int b = blockIdx.x;         // batch index
  const float ci = -avec[N_STATE - 1 - i];

  xbuf[0][i] = 0.0f;                // x_{-1} = 0
  __syncthreads();

  float* __restrict__ p = bu + (size_t)b * SEQ * N_STATE;
  int cur = 0;
  for (int t = 0; t < SEQ; ++t) {
    const float drive = p[(size_t)t * N_STATE + i];        // Bu_t[i]
    const float xm1   = (i == 0) ? 0.0f : xbuf[cur][i - 1];
    const float s     = xbuf[cur][N_STATE - 1];
    const float xn    = fmaf(ci, s, xm1 + drive);
    xbuf[cur ^ 1][i]  = xn;
    p[(size_t)t * N_STATE + i] = xn;                       // xs overwrites Bu
    if (t + 2 < SEQ)
      __builtin_prefetch(&p[(size_t)(t + 2) * N_STATE + i], 0, 1);
    __syncthreads();
    cur ^= 1;
  }
}

// ---------------------------------------------------------------------------
// Phase 3: y[r][p] = sum_n xs[r][n]*C[p][n] + sum_m u[r][m]*D[p][m]
// One wave per 16x16 tile; K = 512 (xs*C) then K = 64 (u*D), same accumulator.
// ---------------------------------------------------------------------------
__global__ __launch_bounds__(256) void ssm_out_gemm(const float* __restrict__ xs,
                                                    const float* __restrict__ u,
                                                    const float* __restrict__ Cmat,
                                                    const float* __restrict__ Dmat,
                                                    float* __restrict__ y) {
  const int wave = (blockIdx.x * 256 + threadIdx.x) >> 5;
  const int lane = threadIdx.x & 31;
  const int half = lane >> 4;
  const int lm   = lane & 15;

  const int colTiles = P_OUT / 16;            // 4
  const int rowTile  = wave / colTiles;       // 0..4095
  const int colTile  = wave % colTiles;       // 0..3

  const int row = rowTile * 16 + lm;
  const int col = colTile * 16 + lm;          // output channel p

  v8f acc = {};

  const float* __restrict__ xrow = xs + (size_t)row * N_STATE;
  const float* __restrict__ crow = Cmat + (size_t)col * N_STATE; // C^T[k][p] = C[p][k]
#pragma unroll 8
  for (int k0 = 0; k0 < N_STATE; k0 += 4) {
    const int ka = k0 + 2 * half;
    v2f av, bv;
    av.x = xrow[ka];
    av.y = xrow[ka + 1];
    bv.x = crow[ka];
    bv.y = crow[ka + 1];
    acc = __builtin_amdgcn_wmma_f32_16x16x4_f32(false, av, false, bv,
                                                (short)0, acc, false, false);
  }

  const float* __restrict__ urow = u + (size_t)row * M_IN;
  const float* __restrict__ drow = Dmat + (size_t)col * M_IN;    // D^T[k][p] = D[p][k]
#pragma unroll
  for (int k0 = 0; k0 < M_IN; k0 += 4) {
    const int ka = k0 + 2 * half;
    v2f av, bv;
    av.x = urow[ka];
    av.y = urow[ka + 1];
    bv.x = drow[ka];
    bv.y = drow[ka + 1];
    acc = __builtin_amdgcn_wmma_f32_16x16x4_f32(false, av, false, bv,
                                                (short)0, acc, false, false);
  }

  float* __restrict__ out = y + (size_t)(rowTile * 16) * P_OUT + colTile * 16 + lm;
#pragma unroll
  for (int v = 0; v < 8; ++v)
    out[(size_t)(v + 8 * half) * P_OUT] = acc[v];
}

// ---------------------------------------------------------------------------
extern "C" void kernel_launch(void* const* d_in, const int* in_sizes, int n_in,
                              void* d_out, int out_size, void* d_ws, size_t ws_size,
                              hipStream_t stream) {
  const float* u = (const float*)d_in[0];   // (32, 2048, 64)
  const float* a = (const float*)d_in[1];   // (512,)
  const float* B = (const float*)d_in[2];   // (512, 64)
  const float* C = (const float*)d_in[3];   // (64, 512)
  const float* D = (const float*)d_in[4];   // (64, 64)
  float* y = (float*)d_out;                 // (32, 2048, 64)

  // Workspace: Bu / xs shared buffer, ROWS * N_STATE fp32 = 128 MiB.
  float* bu = (float*)d_ws;
  (void)in_sizes; (void)n_in; (void)out_size; (void)ws_size;

  // Phase 1: (ROWS/16)*(N_STATE/16) = 131072 waves, 8 waves/block.
  ssm_bu_gemm<<<(ROWS / 16) * (N_STATE / 16) / 8, 256, 0, stream>>>(u, B, bu);

  // Phase 2: one block per batch element.
  ssm_scan<<<BATCH, N_STATE, 0, stream>>>(a, bu);

  // Phase 3: (ROWS/16)*(P_OUT/16) = 16384 waves, 8 waves/block.
  ssm_out_gemm<<<(ROWS / 16) * (P_OUT / 16) / 8, 256, 0, stream>>>(bu, u, C, D, y);
}